// CorefDecoderHOI_48979807043766
// MI455X (gfx1250) — compile-verified
//
#include <hip/hip_runtime.h>
#include <hip/hip_bf16.h>

typedef __attribute__((ext_vector_type(2))) float v2f;
typedef __attribute__((ext_vector_type(8))) float v8f;
typedef unsigned long long u64;

// Only meaningful in the device pass; host pass doesn't know AMDGPU builtins.
#if defined(__HIP_DEVICE_COMPILE__) && !__has_builtin(__builtin_amdgcn_wmma_f32_16x16x4_f32)
#error "missing __builtin_amdgcn_wmma_f32_16x16x4_f32 on this toolchain (device pass)"
#endif

#define NUM_WORDS      5000
#define NUM_CAND       50000
#define NUM_TOP        2000
#define SPAN_DIM       2304
#define UNARY_HDIM     1000
#define SORT_N         65536   // next pow2 >= NUM_CAND
#define NT_PAIRS       32      // 32 pairs of 16-wide N tiles cover 1024 >= 1000
#define M_BLOCK        32      // rows per block (2 M-tiles)
#define M_GRID         1563    // ceil(50000 / 32)

// ---------------------------------------------------------------------------
// Kernel 1: fused  scores = relu(A @ W1 + b1) @ W2 + b2
// Block = 32 rows (2 M-tiles); 4 waves split 32 N-tile pairs.
// 2x2 register blocking: each A frag feeds 2 WMMAs, each B frag feeds 2 WMMAs
// => 1.5 loads per v_wmma_f32_16x16x4_f32 and ~2x less L2 traffic.
// fp32 WMMA layouts per ISA: A 16x4: lanes 0-15 K0/K1 of row=lane, lanes 16-31
// K2/K3 of row=lane-16. B 4x16: v0 = K(k+hi), v1 = K(k+2+hi), N = lane&15.
// C/D: VGPR r -> row r (lanes 0-15) / row r+8 (lanes 16-31).
// ---------------------------------------------------------------------------
__global__ void __launch_bounds__(128)
mention_score_kernel(const float* __restrict__ A, const float* __restrict__ W1,
                     const float* __restrict__ b1, const float* __restrict__ W2,
                     const float* __restrict__ b2, float* __restrict__ scores) {
    __shared__ float sScore[M_BLOCK];
    const int row_base = blockIdx.x * M_BLOCK;
    const int tid  = threadIdx.x;
    const int wave = tid >> 5;
    const int lane = tid & 31;
    if (tid < M_BLOCK) sScore[tid] = 0.f;
    __syncthreads();

    const int hiHalf = (lane >= 16) ? 1 : 0;
    const int mlane  = lane & 15;
    // Last block: clamp row addresses (duplicate row 49999 -> finite garbage,
    // never stored). Epilogue stores only rows < NUM_CAND.
    int rowA0 = row_base + mlane;        if (rowA0 >= NUM_CAND) rowA0 = NUM_CAND - 1;
    int rowA1 = row_base + 16 + mlane;   if (rowA1 >= NUM_CAND) rowA1 = NUM_CAND - 1;
    const float* arow0 = A + (long long)rowA0 * SPAN_DIM + (hiHalf ? 2 : 0);
    const float* arow1 = A + (long long)rowA1 * SPAN_DIM + (hiHalf ? 2 : 0);

    for (int ctp = wave; ctp < NT_PAIRS; ctp += 4) {
        const int n0 = ctp * 32 + mlane;        // first 16-wide N tile
        const int n1 = n0 + 16;                 // second 16-wide N tile
        const int nc0 = (n0 < UNARY_HDIM) ? n0 : (UNARY_HDIM - 1);
        const int nc1 = (n1 < UNARY_HDIM) ? n1 : (UNARY_HDIM - 1);
        const float* b0lo = W1 + nc0 + (hiHalf ? 1 : 0) * UNARY_HDIM;  // K = k + hi
        const float* b0hi = W1 + nc0 + (hiHalf ? 3 : 2) * UNARY_HDIM;  // K = k + 2 + hi
        const float* b1lo = W1 + nc1 + (hiHalf ? 1 : 0) * UNARY_HDIM;
        const float* b1hi = W1 + nc1 + (hiHalf ? 3 : 2) * UNARY_HDIM;

        v8f c00 = {0.f,0.f,0.f,0.f,0.f,0.f,0.f,0.f};
        v8f c01 = c00, c10 = c00, c11 = c00;

        for (int k = 0; k < SPAN_DIM; k += 4) {
            v2f a0 = *(const v2f*)(arow0 + k);   // 8B contiguous per lane
            v2f a1 = *(const v2f*)(arow1 + k);
            const long long ko = (long long)k * UNARY_HDIM;
            v2f bA, bB;
            bA.x = b0lo[ko]; bA.y = b0hi[ko];
            bB.x = b1lo[ko]; bB.y = b1hi[ko];
            c00 = __builtin_amdgcn_wmma_f32_16x16x4_f32(false, a0, false, bA,
                                                        (short)0, c00, false, false);
            c01 = __builtin_amdgcn_wmma_f32_16x16x4_f32(false, a0, false, bB,
                                                        (short)0, c01, false, false);
            c10 = __builtin_amdgcn_wmma_f32_16x16x4_f32(false, a1, false, bA,
                                                        (short)0, c10, false, false);
            c11 = __builtin_amdgcn_wmma_f32_16x16x4_f32(false, a1, false, bB,
                                                        (short)0, c11, false, false);
        }

        // relu + bias + dot with W2, reduced over the 16 N-lanes of each half.
        auto reduce_acc = [&](v8f c, int n, int rowOff) {
            const float bias = (n < UNARY_HDIM) ? b1[n] : 0.f;
            const float wv   = (n < UNARY_HDIM) ? W2[n] : 0.f;
            #pragma unroll
            for (int r = 0; r < 8; ++r) {
                float p = c[r] + bias;
                p = p > 0.f ? p : 0.f;
                p *= wv;
                p += __shfl_xor(p, 1, 32);
                p += __shfl_xor(p, 2, 32);
                p += __shfl_xor(p, 4, 32);
                p += __shfl_xor(p, 8, 32);
                if (lane == 0)       atomicAdd(&sScore[rowOff + r], p);
                else if (lane == 16) atomicAdd(&sScore[rowOff + 8 + r], p);
            }
        };
        reduce_acc(c00, n0, 0);
        reduce_acc(c01, n1, 0);
        reduce_acc(c10, n0, 16);
        reduce_acc(c11, n1, 16);
    }
    __syncthreads();
    if (tid < M_BLOCK) {
        const int row = row_base + tid;
        if (row < NUM_CAND) scores[row] = sScore[tid] + b2[0];
    }
}

// ---------------------------------------------------------------------------
// Sort 1: key = (~monotonic(score) << 32) | idx  => ascending sort gives
// descending score, ties broken by ascending idx (stable-argsort semantics).
// ---------------------------------------------------------------------------
__global__ void build_keys1_kernel(const float* __restrict__ scores, u64* __restrict__ keys) {
    int i = blockIdx.x * blockDim.x + threadIdx.x;
    if (i >= SORT_N) return;
    u64 key;
    if (i < NUM_CAND) {
        unsigned u = __float_as_uint(scores[i]);
        unsigned m = (u & 0x80000000u) ? ~u : (u | 0x80000000u);  // order-preserving
        unsigned d = ~m;                                           // descending
        key = ((u64)d << 32) | (unsigned)i;
    } else {
        key = ((u64)0xFFFFFFFFu << 32) | (unsigned)i;              // pad to the back
    }
    keys[i] = key;
}

__global__ void bitonic_stage_kernel(u64* __restrict__ keys, int j, int k) {
    int i = blockIdx.x * blockDim.x + threadIdx.x;
    int ixj = i ^ j;
    if (ixj <= i) return;
    u64 a = keys[i], b = keys[ixj];
    bool up = ((i & k) == 0);
    if ((a > b) == up) { keys[i] = b; keys[ixj] = a; }
}

// ---------------------------------------------------------------------------
// Sequential crossing-bracket scan (faithful port of extract_top_spans_hoi).
// Single workgroup; dicts live in LDS; the <=11 overlap checks are lane-parallel.
// ---------------------------------------------------------------------------
__global__ void extract_spans_kernel(const u64* __restrict__ keys1,
                                     const int* __restrict__ starts,
                                     const int* __restrict__ ends,
                                     int* __restrict__ accepts,
                                     int* __restrict__ dcount,
                                     const int* __restrict__ ntsp) {
    __shared__ int stme[NUM_WORDS];   // start -> max end
    __shared__ int etms[NUM_WORDS];   // end   -> min start
    __shared__ int scount;
    const int tid = threadIdx.x;
    for (int i = tid; i < NUM_WORDS; i += blockDim.x) { stme[i] = -1; etms[i] = -1; }
    if (tid == 0) scount = 0;
    __syncthreads();
    const int nts = ntsp[0];

    for (int cand = 0; cand < NUM_CAND; ++cand) {
        const int idx = (int)(keys1[cand] & 0xFFFFFFFFull);
        const int s = starts[idx];
        const int e = ends[idx];
        int cross = 0;
        if (tid <= 10) {
            int pos   = s + tid;
            int posc  = pos > (NUM_WORDS - 1) ? (NUM_WORDS - 1) : pos;
            bool valid = (pos <= e);
            int me = stme[posc];
            int ms = etms[posc];
            if (valid && tid > 0 && me > e)                cross = 1;
            if (valid && pos < e && ms >= 0 && ms < s)     cross = 1;
        }
        int any = __syncthreads_or(cross);
        if (tid == 0) {
            int accept = (!any) && (scount < nts);
            if (accept) {
                if (e > stme[s]) stme[s] = e;
                int cur = etms[e];
                if (cur == -1 || s < cur) etms[e] = s;
                scount += 1;
            }
            accepts[cand] = accept;
        }
        __syncthreads();
    }
    if (tid == 0) dcount[0] = scount;
}

// ---------------------------------------------------------------------------
// Sort 2: order accepted spans by (start*NUM_WORDS+end); position packed into
// the low 16 bits provides jnp-stable tie-breaking; rejected/pad go to back.
// ---------------------------------------------------------------------------
__global__ void build_keys2_kernel(const u64* __restrict__ keys1,
                                   const int* __restrict__ accepts,
                                   const int* __restrict__ starts,
                                   const int* __restrict__ ends,
                                   u64* __restrict__ keys2) {
    int i = blockIdx.x * blockDim.x + threadIdx.x;
    if (i >= SORT_N) return;
    u64 key;
    if (i < NUM_CAND) {
        int idx = (int)(keys1[i] & 0xFFFFFFFFull);
        if (accepts[i]) {
            u64 ok = (u64)(starts[idx] * NUM_WORDS + ends[idx]);   // < 2^25
            key = (ok << 16) | (u64)(unsigned)i;
        } else {
            key = ((u64)0xFFFFFFFFu << 16) | (u64)(unsigned)i;
        }
    } else {
        key = 0xFFFF000000000000ull | (u64)(unsigned)i;
    }
    keys2[i] = key;
}

// ---------------------------------------------------------------------------
// Gather: out = [top_idx (2000) | top_scores (2000) | top_emb (2000x2304)]
// ---------------------------------------------------------------------------
__global__ void gather_kernel(const u64* __restrict__ keys1,
                              const u64* __restrict__ keys2,
                              const int* __restrict__ dcount,
                              const float* __restrict__ scores,
                              const float* __restrict__ emb,
                              float* __restrict__ out) {
    const int i = blockIdx.x;              // 0..NUM_TOP-1
    const int count = dcount[0];
    const int use = (i < count) ? i : 0;   // pad with first selected span
    int pos = (int)(keys2[use] & 0xFFFFull);
    int idx = (int)(keys1[pos] & 0xFFFFFFFFull);
    if (idx >= NUM_CAND) idx = NUM_CAND - 1;   // degenerate safety clamp
    if (threadIdx.x == 0) {
        out[i]            = (float)idx;
        out[NUM_TOP + i]  = scores[idx];
    }
    const float* src = emb + (long long)idx * SPAN_DIM;
    float* dst = out + 2 * NUM_TOP + (long long)i * SPAN_DIM;
    for (int c = threadIdx.x; c < SPAN_DIM; c += blockDim.x) dst[c] = src[c];
}

// ---------------------------------------------------------------------------
extern "C" void kernel_launch(void* const* d_in, const int* in_sizes, int n_in,
                              void* d_out, int out_size, void* d_ws, size_t ws_size,
                              hipStream_t stream) {
    const float* span_emb = (const float*)d_in[0];
    const float* W1       = (const float*)d_in[1];
    const float* b1       = (const float*)d_in[2];
    const float* W2       = (const float*)d_in[3];
    const float* b2       = (const float*)d_in[4];
    const int*   starts   = (const int*)d_in[5];
    const int*   ends     = (const int*)d_in[6];
    const int*   nts      = (const int*)d_in[7];
    float* out = (float*)d_out;

    char* ws = (char*)d_ws;
    float* scores  = (float*)(ws);                               // 200 KB
    u64*   keys1   = (u64*)(ws + 262144);                        // 512 KB
    u64*   keys2   = (u64*)(ws + 262144 + 524288);               // 512 KB
    int*   accepts = (int*)(ws + 262144 + 2 * 524288);           // 200 KB
    int*   dcount  = (int*)(ws + 262144 + 2 * 524288 + 204800);  // 4 B

    // 1) fused mention-score GEMM (fp32 WMMA, 2x2 register blocking)
    mention_score_kernel<<<M_GRID, 128, 0, stream>>>(span_emb, W1, b1, W2, b2, scores);

    // 2) stable argsort by descending score (bitonic on packed 64-bit keys)
    build_keys1_kernel<<<SORT_N / 256, 256, 0, stream>>>(scores, keys1);
    for (int k = 2; k <= SORT_N; k <<= 1)
        for (int j = k >> 1; j > 0; j >>= 1)
            bitonic_stage_kernel<<<SORT_N / 256, 256, 0, stream>>>(keys1, j, k);

    // 3) sequential crossing-span filter
    extract_spans_kernel<<<1, 64, 0, stream>>>(keys1, starts, ends, accepts, dcount, nts);

    // 4) stable re-order by (start,end), accepted first
    build_keys2_kernel<<<SORT_N / 256, 256, 0, stream>>>(keys1, accepts, starts, ends, keys2);
    for (int k = 2; k <= SORT_N; k <<= 1)
        for (int j = k >> 1; j > 0; j >>= 1)
            bitonic_stage_kernel<<<SORT_N / 256, 256, 0, stream>>>(keys2, j, k);

    // 5) gather outputs
    gather_kernel<<<NUM_TOP, 256, 0, stream>>>(keys1, keys2, dcount, scores, span_emb, out);

    (void)in_sizes; (void)n_in; (void)out_size; (void)ws_size;
}